// FixedRadiusNearNeighbors_33698313404548
// MI455X (gfx1250) — compile-verified
//
#include <hip/hip_runtime.h>

typedef __attribute__((ext_vector_type(2))) float        v2f;
typedef __attribute__((ext_vector_type(8))) float        v8f;
typedef __attribute__((ext_vector_type(4))) unsigned int v4u;
typedef __attribute__((ext_vector_type(4))) int          v4i;
typedef __attribute__((ext_vector_type(8))) int          v8i;

#define RADIUS2      (0.1f * 0.1f)
#define NPTS         16384
#define NCENT        1024
#define NNEI         64
#define WAVES_PER_WG 4
#define CHUNKPTS     2048
#define NCHUNK       (NPTS / CHUNKPTS)   // 8
#define CHUNKELEMS   (CHUNKPTS * 3)      // 6144 f32
#define CHUNKBYTES   (CHUNKELEMS * 4)    // 24 KB per buffer
#define ITERS        (CHUNKPTS / 16)     // 128 WMMA iters per chunk

// ---- Tensor Data Mover: DMA one contiguous 24KB chunk of pos into LDS ----
// D# per CDNA5 ISA §8.3/8.4: group0 = {count=1, lds_addr, global_addr, type=2},
// group1 = {data_size=4B, tensor_dim0=tile_dim0=stride0=6144, tensor_dim1=1}.
__device__ __forceinline__ void tdm_load_chunk(const float* gsrc, unsigned ldsAddr) {
    const unsigned long long ga = (unsigned long long)(uintptr_t)gsrc;
    v4u g0;
    g0[0] = 1u;                                               // count=1 (valid user D#)
    g0[1] = ldsAddr;                                          // LDS byte address
    g0[2] = (unsigned)ga;                                     // global_addr[31:0]
    g0[3] = (unsigned)((ga >> 32) & 0x01FFFFFFull)            // global_addr[56:32]
          | 0x80000000u;                                      // type=2 ("image")
    v8i g1;
    g1[0] = 0x00020000;                                       // data_size=2 (4 bytes)
    g1[1] = (int)(((unsigned)CHUNKELEMS & 0xFFFFu) << 16);    // tensor_dim0[15:0]
    g1[2] = (int)((((unsigned)CHUNKELEMS >> 16) & 0xFFFFu)    // tensor_dim0[31:16]
          | (1u << 16));                                      // tensor_dim1 = 1
    g1[3] = (int)((unsigned)CHUNKELEMS << 16);                // tile_dim0 = 6144
    g1[4] = 0;                                                // tile_dim1/2 unused
    g1[5] = CHUNKELEMS;                                       // tensor_dim0_stride
    g1[6] = 0;
    g1[7] = 0;
    v4i z4 = {0, 0, 0, 0};
#if __clang_major__ >= 23
    v8i z8 = {0, 0, 0, 0, 0, 0, 0, 0};
    __builtin_amdgcn_tensor_load_to_lds(g0, g1, z4, z4, z8, 0);
#else
    __builtin_amdgcn_tensor_load_to_lds(g0, g1, z4, z4, 0);
#endif
}

// V_WRITELANE_B32 (no clang builtin on this toolchain): put wave-uniform
// SGPR value into one lane of a VGPR. Never skipped by EXEC per ISA.
#define WRITELANE(dstv, sval, lane) \
    asm("v_writelane_b32 %0, %1, " #lane : "+v"(dstv) : "s"(sval))

// One wave = one tile of 16 centroids vs all 16384 points of its batch.
// 4 waves/WG share TDM-staged, double-buffered point chunks in LDS.
// Distances 16x16 at a time with V_WMMA_F32_16X16X4_F32:
//   A(16x4)=[-2cx,-2cy,-2cz,1], B(4x16)=[px,py,pz,|p|^2], C=|c|^2 => D=|c-p|^2
__global__ __launch_bounds__(32 * WAVES_PER_WG)
void FixedRadiusNearNeighbors_wmma_kernel(const float* __restrict__ pos,
                                          const int*   __restrict__ cent,
                                          int*         __restrict__ out)
{
    extern __shared__ char smem[];                 // 2 x 24 KB buffers (+pad)

    const unsigned l      = threadIdx.x & 31;      // lane
    const unsigned waveId = threadIdx.x >> 5;      // 0..3
    const unsigned tile   = blockIdx.x * WAVES_PER_WG + waveId;  // 0..511
    const unsigned b      = tile >> 6;             // batch (same for all 4 waves)
    const unsigned stile  = tile & 63;
    const unsigned m      = l & 15;                // centroid row this lane mirrors
    const bool     lo     = (l < 16);

    const float* posB = pos + (size_t)b * NPTS * 3;
    const int    s    = (int)(stile * 16 + m);
    const int    cidx = cent[b * NCENT + s];

    const float cx  = posB[(size_t)cidx * 3 + 0];
    const float cy  = posB[(size_t)cidx * 3 + 1];
    const float cz  = posB[(size_t)cidx * 3 + 2];
    const float csq = cx * cx + cy * cy + cz * cz;

    // A fragment (−2 folded here, built once):
    // VGPR0 = K0 (lanes 0-15) / K2 (lanes 16-31); VGPR1 = K1 / K3
    v2f a;
    a[0] = -2.0f * (lo ? cx : cz);
    a[1] = lo ? (-2.0f * cy) : 1.0f;

    // C fragment: VGPR i = csq[row i] (lanes 0-15), csq[row i+8] (lanes 16-31)
    v8f c;
    const int csqBits = __builtin_bit_cast(int, csq);
    #pragma unroll
    for (int i = 0; i < 8; ++i) {
        const int loBits = __builtin_amdgcn_readlane(csqBits, i);
        const int hiBits = __builtin_amdgcn_readlane(csqBits, i + 8);
        c[i] = __builtin_bit_cast(float, lo ? loBits : hiBits);
    }

    int count = lo ? 0 : NNEI;                     // upper-half lanes own no row
    int first = cidx;
    const size_t outRow = ((size_t)b * NCENT + (size_t)stile * 16 + m) * NNEI;
    const int bpIdx = (int)((m & 7) << 2);         // ds_bpermute byte index

    // ---- prologue: stage chunk 0 ----
    if (waveId == 0) {
        tdm_load_chunk(posB, (unsigned)(uintptr_t)smem);
        __builtin_amdgcn_s_wait_tensorcnt(0);
    }
    __syncthreads();

    for (int ch = 0; ch < NCHUNK; ++ch) {
        // overlap: DMA next chunk while computing on current one
        if (waveId == 0 && ch + 1 < NCHUNK) {
            tdm_load_chunk(posB + (size_t)(ch + 1) * CHUNKELEMS,
                           (unsigned)(uintptr_t)(smem + ((ch + 1) & 1) * CHUNKBYTES));
        }

        const float* buf = (const float*)(smem + (ch & 1) * CHUNKBYTES);

        // register-pipelined LDS reads, strength-reduced pointer (+192 B/iter).
        // Final prefetch runs 192 B past the buffer: stays inside the padded
        // LDS allocation and the value is discarded.
        const float* pp = buf + m * 3;
        float px = pp[0];
        float py = pp[1];
        float pz = pp[2];

        for (int i = 0; i < ITERS; ++i) {
            pp += 48;
            const float nx = pp[0];
            const float ny = pp[1];
            const float nz = pp[2];

            const float psq = px * px + py * py + pz * pz;
            v2f bf;
            bf[0] = lo ? px : pz;
            bf[1] = lo ? py : psq;

            v8f d = __builtin_amdgcn_wmma_f32_16x16x4_f32(
                false, a, false, bf, (short)0, c, false, false);

            // cheap any-hit test: min-tree + single ballot
            const float mn = fminf(fminf(fminf(d[0], d[1]), fminf(d[2], d[3])),
                                   fminf(fminf(d[4], d[5]), fminf(d[6], d[7])));
            if (__builtin_amdgcn_ballot_w32(mn <= RADIUS2)) {
                // pack the 8 row ballots into lanes 0..7 via v_writelane,
                // then each lane grabs its row's mask with one ds_bpermute
                const unsigned b0 = __builtin_amdgcn_ballot_w32(d[0] <= RADIUS2);
                const unsigned b1 = __builtin_amdgcn_ballot_w32(d[1] <= RADIUS2);
                const unsigned b2 = __builtin_amdgcn_ballot_w32(d[2] <= RADIUS2);
                const unsigned b3 = __builtin_amdgcn_ballot_w32(d[3] <= RADIUS2);
                const unsigned b4 = __builtin_amdgcn_ballot_w32(d[4] <= RADIUS2);
                const unsigned b5 = __builtin_amdgcn_ballot_w32(d[5] <= RADIUS2);
                const unsigned b6 = __builtin_amdgcn_ballot_w32(d[6] <= RADIUS2);
                const unsigned b7 = __builtin_amdgcn_ballot_w32(d[7] <= RADIUS2);
                int mv = 0;
                WRITELANE(mv, b0, 0);
                WRITELANE(mv, b1, 1);
                WRITELANE(mv, b2, 2);
                WRITELANE(mv, b3, 3);
                WRITELANE(mv, b4, 4);
                WRITELANE(mv, b5, 5);
                WRITELANE(mv, b6, 6);
                WRITELANE(mv, b7, 7);
                const unsigned mm = (unsigned)__builtin_amdgcn_ds_bpermute(bpIdx, mv);
                unsigned rowmask = lo ? ((m < 8) ? (mm & 0xFFFFu) : (mm >> 16)) : 0u;
                const int base = ch * CHUNKPTS + i * 16;
                while (rowmask && count < NNEI) {
                    const int bit = __builtin_ctz(rowmask);
                    const int idx = base + bit;
                    if (count == 0) first = idx;
                    out[outRow + count] = idx;
                    ++count;
                    rowmask &= rowmask - 1;
                }
            }

            px = nx; py = ny; pz = nz;
        }

        if (waveId == 0 && ch + 1 < NCHUNK)
            __builtin_amdgcn_s_wait_tensorcnt(0);
        __syncthreads();
    }

    if (lo) {
        for (int k = count; k < NNEI; ++k) out[outRow + k] = first;
    }
}

extern "C" void kernel_launch(void* const* d_in, const int* in_sizes, int n_in,
                              void* d_out, int out_size, void* d_ws, size_t ws_size,
                              hipStream_t stream) {
    const float* pos  = (const float*)d_in[0];   // (8, 16384, 3) f32
    const int*   cent = (const int*)d_in[1];     // (8, 1024) i32
    int*         out  = (int*)d_out;             // (8, 1024, 64) i32

    const int ntiles  = 8 * (NCENT / 16);            // 512 waves total
    const int nblocks = ntiles / WAVES_PER_WG;       // 128 workgroups
    const size_t shmem = 2 * CHUNKBYTES + 256;       // double buffer + OOB pad

    hipLaunchKernelGGL(FixedRadiusNearNeighbors_wmma_kernel,
                       dim3(nblocks), dim3(32 * WAVES_PER_WG), shmem, stream,
                       pos, cent, out);
}